// MixHop_6828998001548
// MI455X (gfx1250) — compile-verified
//
#include <hip/hip_runtime.h>
#include <hip/hip_bf16.h>

#define IN_C   128
#define HID    96
#define OUTC   64
#define W0CAT  (HID * 3)    // 288
#define W1CAT  (OUTC * 3)   // 192
#define BN_EPS 1e-5f

typedef float v2f __attribute__((ext_vector_type(2)));
typedef float v8f __attribute__((ext_vector_type(8)));

// ---------------- utility kernels ----------------

__global__ void zero_f32(float* __restrict__ p, long long count) {
  long long i = (long long)blockIdx.x * blockDim.x + threadIdx.x;
  long long stride = (long long)gridDim.x * blockDim.x;
  for (; i < count; i += stride) p[i] = 0.0f;
}

__global__ void init_deg(float* __restrict__ deg, int n) {
  int i = blockIdx.x * blockDim.x + threadIdx.x;
  if (i < n) deg[i] = 1.0f;   // self-loop contribution
}

__global__ void count_deg(const int* __restrict__ cols, float* __restrict__ deg, int e) {
  int i = blockIdx.x * blockDim.x + threadIdx.x;
  if (i < e) atomicAdd(&deg[cols[i]], 1.0f);
}

__global__ void finish_dinv(float* __restrict__ deg, int n) {
  int i = blockIdx.x * blockDim.x + threadIdx.x;
  if (i < n) {
    float d = deg[i];
    deg[i] = (d > 0.0f) ? rsqrtf(d) : 0.0f;   // deg >= 1 always, but match ref
  }
}

// ---------------- fp32 WMMA GEMM: C[m,n] = A[m,:] @ W[:,n] + bias[n] ----------------
// One wave computes one 16 x (16*NT) output strip: the A fragment is loaded once
// per K-step and reused across NT V_WMMA_F32_16X16X4_F32 issues (exact fp32).
// Requires M % 16 == 0, K % 4 == 0; W is (K x 16*NT) row-major with compile-time
// row stride so B fragment loads become immediate-offset clauses.
template <int NT>
__global__ __launch_bounds__(128)
void gemm_wmma_f32(const float* __restrict__ A, int lda,
                   const float* __restrict__ W,
                   const float* __restrict__ bias,
                   float* __restrict__ C, int ldc,
                   int M, int K) {
  constexpr int LDW = 16 * NT;           // W row stride (elements)
  const int lane = threadIdx.x & 31;
  const int wave = threadIdx.x >> 5;
  const int m0 = (blockIdx.x * 4 + wave) * 16;
  if (m0 >= M) return;                   // wave-uniform: EXEC all-ones for WMMA

  const int mr = lane & 15;              // row within tile (A) / col within tile (B,C)
  const int kh = lane >> 4;              // K-half select (0/1)

  v8f acc[NT];
#pragma unroll
  for (int t = 0; t < NT; ++t) acc[t] = {};

  const float* ap = A + (long long)(m0 + mr) * lda + 2 * kh;  // A[m0+mr][k0 + 2*kh]
  const float* bp = W + (long long)(2 * kh) * LDW + mr;       // W[k0 + 2*kh][mr]

  for (int k0 = 0; k0 < K; k0 += 4) {
    v2f a;
    a.x = ap[0];                         // K = k0 + 2*kh
    a.y = ap[1];                         // K = k0 + 2*kh + 1
    ap += 4;

    v2f b[NT];
#pragma unroll
    for (int t = 0; t < NT; ++t) {       // immediate-offset loads -> s_clause
      b[t].x = bp[16 * t];               // W[k0+2*kh  ][16*t + mr]
      b[t].y = bp[LDW + 16 * t];         // W[k0+2*kh+1][16*t + mr]
    }
    bp += 4 * LDW;

#pragma unroll
    for (int t = 0; t < NT; ++t)
      acc[t] = __builtin_amdgcn_wmma_f32_16x16x4_f32(
          /*neg_a=*/false, a, /*neg_b=*/false, b[t],
          /*c_mod=*/(short)0, acc[t], /*reuse_a=*/false, /*reuse_b=*/false);
  }

  float* cp = C + (long long)(m0 + 8 * kh) * ldc + mr;
#pragma unroll
  for (int t = 0; t < NT; ++t) {
    const float bcol = bias[16 * t + mr];
#pragma unroll
    for (int r = 0; r < 8; ++r)          // row = m0 + 8*kh + r, col = 16*t + mr
      cp[(long long)r * ldc + 16 * t] = acc[t][r] + bcol;
  }
}

// ---------------- SpMM scatter: Y[c,:] += dinv[r]*dinv[c] * X[r,:] ----------------
// Logical edge list = real edges [0,e) plus self-loops [e, e+n).
__global__ void spmm_atomic(const int* __restrict__ rows, const int* __restrict__ cols,
                            const float* __restrict__ dinv,
                            const float* __restrict__ X, int ldx,
                            float* __restrict__ Y, int ldy,
                            int e, int n) {
  int i = blockIdx.x * blockDim.x + threadIdx.x;  // edge id
  int c4 = blockIdx.y * 4;                        // feature offset
  if (i >= e + n) return;
  int r, c;
  if (i < e) { r = rows[i]; c = cols[i]; }
  else       { r = c = i - e; }
  float wv = dinv[r] * dinv[c];
  const float* xp = X + (long long)r * ldx + c4;
  float*       yp = Y + (long long)c * ldy + c4;
  atomicAdd(yp + 0, wv * xp[0]);
  atomicAdd(yp + 1, wv * xp[1]);
  atomicAdd(yp + 2, wv * xp[2]);
  atomicAdd(yp + 3, wv * xp[3]);
}

// ---------------- fused eval-BatchNorm + ReLU ----------------
__global__ void bn_relu(float* __restrict__ h,
                        const float* __restrict__ gamma, const float* __restrict__ beta,
                        const float* __restrict__ mean,  const float* __restrict__ var,
                        long long total, int width) {
  long long i = (long long)blockIdx.x * blockDim.x + threadIdx.x;
  long long stride = (long long)gridDim.x * blockDim.x;
  for (; i < total; i += stride) {
    int c = (int)(i % width);
    float v = h[i];
    v = gamma[c] * (v - mean[c]) * rsqrtf(var[c] + BN_EPS) + beta[c];
    h[i] = v > 0.0f ? v : 0.0f;
  }
}

// ---------------- host-side orchestration ----------------

static inline void launch_gemm(const float* A, int lda, const float* W,
                               const float* bias, float* C, int ldc,
                               int M, int K, int N, hipStream_t stream) {
  dim3 grid(((M + 15) / 16 + 3) / 4);
  if (N == HID)
    gemm_wmma_f32<HID / 16><<<grid, 128, 0, stream>>>(A, lda, W, bias, C, ldc, M, K);
  else
    gemm_wmma_f32<OUTC / 16><<<grid, 128, 0, stream>>>(A, lda, W, bias, C, ldc, M, K);
}

static inline void launch_spmm(const int* rows, const int* cols, const float* dinv,
                               const float* X, int ldx, float* Y, int ldy,
                               int e, int n, int width, hipStream_t stream) {
  dim3 grid((e + n + 127) / 128, width / 4);
  spmm_atomic<<<grid, 128, 0, stream>>>(rows, cols, dinv, X, ldx, Y, ldy, e, n);
}

extern "C" void kernel_launch(void* const* d_in, const int* in_sizes, int n_in,
                              void* d_out, int out_size, void* d_ws, size_t ws_size,
                              hipStream_t stream) {
  const float* x        = (const float*)d_in[0];
  const int*   eidx     = (const int*)d_in[1];
  const float* conv0_W  = (const float*)d_in[2];   // (3,128,96)
  const float* conv0_b  = (const float*)d_in[3];   // (3,96)
  const float* conv1_W  = (const float*)d_in[4];   // (3,288,64)
  const float* conv1_b  = (const float*)d_in[5];   // (3,64)
  const float* bn_gamma = (const float*)d_in[6];
  const float* bn_beta  = (const float*)d_in[7];
  const float* bn_mean  = (const float*)d_in[8];
  const float* bn_var   = (const float*)d_in[9];
  const float* fp_W     = (const float*)d_in[10];  // (192,64)
  const float* fp_b     = (const float*)d_in[11];  // (64,)
  float* out = (float*)d_out;

  const int n = in_sizes[0] / IN_C;
  const int e = in_sizes[1] / 2;
  const int* rows = eidx;
  const int* cols = eidx + e;

  // workspace partition (all fp32, 256B-aligned chunks)
  float* ws = (float*)d_ws;
  size_t off = 0;
  auto alloc = [&](size_t cnt) { float* p = ws + off; off += (cnt + 63) & ~(size_t)63; return p; };
  float* deg = alloc((size_t)n);              // becomes dinv in place
  float* h0  = alloc((size_t)n * W0CAT);      // layer0 concat (n x 288)
  float* h1  = alloc((size_t)n * W1CAT);      // layer1 concat (n x 192)
  float* ta  = alloc((size_t)n * HID);        // temp (n x 96, reused as n x 64)
  float* tb  = alloc((size_t)n * HID);        // temp (n x 96, reused as n x 64)

  // --- gcn_norm: deg -> dinv ---
  init_deg<<<(n + 255) / 256, 256, 0, stream>>>(deg, n);
  count_deg<<<(e + 255) / 256, 256, 0, stream>>>(cols, deg, e);
  finish_dinv<<<(n + 255) / 256, 256, 0, stream>>>(deg, n);

  // --- MixHop layer 0: x(128) -> h0(288) ---
  zero_f32<<<4096, 256, 0, stream>>>(h0, (long long)n * W0CAT);
  // hop 0: assign into cols [0,96)
  launch_gemm(x, IN_C, conv0_W + 0 * IN_C * HID, conv0_b + 0 * HID,
              h0 + 0, W0CAT, n, IN_C, HID, stream);
  // hop 1: gemm -> ta, spmm -> cols [96,192)
  launch_gemm(x, IN_C, conv0_W + 1 * IN_C * HID, conv0_b + 1 * HID,
              ta, HID, n, IN_C, HID, stream);
  launch_spmm(rows, cols, deg, ta, HID, h0 + HID, W0CAT, e, n, HID, stream);
  // hop 2: gemm -> ta, spmm -> tb, spmm -> cols [192,288)
  launch_gemm(x, IN_C, conv0_W + 2 * IN_C * HID, conv0_b + 2 * HID,
              ta, HID, n, IN_C, HID, stream);
  zero_f32<<<4096, 256, 0, stream>>>(tb, (long long)n * HID);
  launch_spmm(rows, cols, deg, ta, HID, tb, HID, e, n, HID, stream);
  launch_spmm(rows, cols, deg, tb, HID, h0 + 2 * HID, W0CAT, e, n, HID, stream);

  // --- BatchNorm(eval) + ReLU in place ---
  bn_relu<<<4096, 256, 0, stream>>>(h0, bn_gamma, bn_beta, bn_mean, bn_var,
                                    (long long)n * W0CAT, W0CAT);

  // --- MixHop layer 1: h0(288) -> h1(192) ---
  zero_f32<<<4096, 256, 0, stream>>>(h1, (long long)n * W1CAT);
  launch_gemm(h0, W0CAT, conv1_W + 0 * W0CAT * OUTC, conv1_b + 0 * OUTC,
              h1 + 0, W1CAT, n, W0CAT, OUTC, stream);
  launch_gemm(h0, W0CAT, conv1_W + 1 * W0CAT * OUTC, conv1_b + 1 * OUTC,
              ta, OUTC, n, W0CAT, OUTC, stream);
  launch_spmm(rows, cols, deg, ta, OUTC, h1 + OUTC, W1CAT, e, n, OUTC, stream);
  launch_gemm(h0, W0CAT, conv1_W + 2 * W0CAT * OUTC, conv1_b + 2 * OUTC,
              ta, OUTC, n, W0CAT, OUTC, stream);
  zero_f32<<<4096, 256, 0, stream>>>(tb, (long long)n * OUTC);
  launch_spmm(rows, cols, deg, ta, OUTC, tb, OUTC, e, n, OUTC, stream);
  launch_spmm(rows, cols, deg, tb, OUTC, h1 + 2 * OUTC, W1CAT, e, n, OUTC, stream);

  // --- final projection: h1(192) @ fp_W(192x64) + fp_b -> out(n x 64) ---
  launch_gemm(h1, W1CAT, fp_W, fp_b, out, OUTC, n, W1CAT, OUTC, stream);
}